// GINNet_WITH_EMBEDDINGTianshouPPO_ACTOR_44976897524029
// MI455X (gfx1250) — compile-verified
//
#include <hip/hip_runtime.h>
#include <hip/hip_bf16.h>
#include <math.h>

// ---------------------------------------------------------------------------
// GAT(3 layers) + aisle scatter-mean + MLP head + masked softmax for MI455X
// (gfx1250). GEMMs use V_WMMA_F32_16X16X4_F32 (full f32 precision: the final
// softmax amplifies logit error, so no f16 downcast). Sparse segment ops use
// native CDNA5 f32/u32 global atomics. Everything fits in the 192MB L2.
// ---------------------------------------------------------------------------

typedef __attribute__((ext_vector_type(2))) float v2f;
typedef __attribute__((ext_vector_type(8))) float v8f;

#define NEG_SLOPE_GAT 0.2f
#define NEG_SLOPE_MLP 0.01f
#define ORD_NEG_INF   0x007FFFFFu   // order(-inf) under monotone float->uint map

// ============================ WMMA GEMM ====================================
// C[M,Nc] = act(A[M,K] @ W[K,Nc] + bias), one wave per 16x16 C tile.
// A-matrix VGPR layout (32-bit, 16x4): lane&15 = M row; lane>=16 holds K+2/K+3.
// B-matrix mirrored: lane&15 = N col; half-wave selects K pair.
// C/D: VGPR r -> M = r + 8*(lane>=16), N = lane&15.
__global__ void k_gemm_wmma(const float* __restrict__ A, const float* __restrict__ W,
                            const float* __restrict__ bias, float* __restrict__ C,
                            int M, int K, int Nc, float slope, int act)
{
    const int wavesPerBlock = blockDim.x >> 5;
    const int tile   = blockIdx.x * wavesPerBlock + ((int)threadIdx.x >> 5);
    const int tilesN = Nc >> 4;
    if (tile >= (M >> 4) * tilesN) return;      // never taken: counts divide evenly
    const int tm   = (tile / tilesN) << 4;
    const int tn   = (tile % tilesN) << 4;
    const int lane = threadIdx.x & 31;
    const int half = lane >> 4;                 // 0: K,K+1   1: K+2,K+3
    const int l    = lane & 15;

    v8f acc = {0.f, 0.f, 0.f, 0.f, 0.f, 0.f, 0.f, 0.f};
    const float* arow = A + (size_t)(tm + l) * K;
    for (int k0 = 0; k0 < K; k0 += 4) {
        const int ka = k0 + half * 2;
        v2f av, bv;
        av.x = arow[ka];
        av.y = arow[ka + 1];
        const float* wp = W + (size_t)ka * Nc + tn + l;
        bv.x = wp[0];
        bv.y = wp[Nc];
        acc = __builtin_amdgcn_wmma_f32_16x16x4_f32(false, av, false, bv,
                                                    (short)0, acc, false, false);
    }
    const int   col = tn + l;
    const float bb  = bias ? bias[col] : 0.0f;
    #pragma unroll
    for (int r = 0; r < 8; ++r) {
        float v = acc[r] + bb;
        if (act) v = (v >= 0.f) ? v : slope * v;
        C[(size_t)(tm + half * 8 + r) * Nc + col] = v;
    }
}

// ===================== per-node dual dot product (es, ed) ==================
__global__ void k_node_dots(const float* __restrict__ H, const float* __restrict__ asrc,
                            const float* __restrict__ adst, float* __restrict__ es,
                            float* __restrict__ ed, int NT, int C)
{
    const int node = blockIdx.x * (blockDim.x >> 5) + ((int)threadIdx.x >> 5);
    const int lane = threadIdx.x & 31;
    if (node >= NT) return;
    float s = 0.f, d = 0.f;
    const float* h = H + (size_t)node * C;
    for (int c = lane; c < C; c += 32) {
        const float hv = h[c];
        s += hv * asrc[c];
        d += hv * adst[c];
    }
    #pragma unroll
    for (int off = 16; off; off >>= 1) {
        s += __shfl_down(s, off, 32);
        d += __shfl_down(d, off, 32);
    }
    if (lane == 0) { es[node] = s; ed[node] = d; }
}

// ============================ edge helpers =================================
__global__ void k_build_edges(const int* __restrict__ links, int* __restrict__ src,
                              int* __restrict__ dst, int B, int E, int N, int ET)
{
    const int i = blockIdx.x * blockDim.x + threadIdx.x;
    if (i >= ET) return;
    const int BE = B * E;
    if (i < BE) {
        const int b = i / E, e = i % E;
        const size_t base = (size_t)b * 2 * E;
        src[i] = links[base + e] + b * N;
        dst[i] = links[base + E + e] + b * N;
    } else {
        const int n = i - BE;   // self loops
        src[i] = n; dst[i] = n;
    }
}

__global__ void k_init_mz(unsigned* __restrict__ mord, float* __restrict__ z, int NT)
{
    const int i = blockIdx.x * blockDim.x + threadIdx.x;
    if (i < NT) { mord[i] = ORD_NEG_INF; z[i] = 0.f; }
}

__global__ void k_init_bias(float* __restrict__ out, const float* __restrict__ bias,
                            int NT, int C)
{
    const long long idx = (long long)blockIdx.x * blockDim.x + threadIdx.x;
    if (idx < (long long)NT * C) out[idx] = bias[(int)(idx % C)];
}

__device__ __forceinline__ unsigned f32_order(float f) {
    unsigned u = __float_as_uint(f);
    return (u >> 31) ? ~u : (u | 0x80000000u);
}
__device__ __forceinline__ float f32_unorder(unsigned o) {
    unsigned u = (o & 0x80000000u) ? (o & 0x7FFFFFFFu) : ~o;
    return __uint_as_float(u);
}

__global__ void k_edge_scores(const float* __restrict__ es, const float* __restrict__ ed,
                              const int* __restrict__ src, const int* __restrict__ dst,
                              float* __restrict__ ebuf, unsigned* __restrict__ mord, int ET)
{
    const int i = blockIdx.x * blockDim.x + threadIdx.x;
    if (i >= ET) return;
    float e = es[src[i]] + ed[dst[i]];
    e = (e >= 0.f) ? e : NEG_SLOPE_GAT * e;
    ebuf[i] = e;
    atomicMax(&mord[dst[i]], f32_order(e));   // segment max via ordered-uint atomic
}

__global__ void k_edge_w(const float* __restrict__ ebuf, const unsigned* __restrict__ mord,
                         const int* __restrict__ dst, float* __restrict__ wbuf,
                         float* __restrict__ z, int ET)
{
    const int i = blockIdx.x * blockDim.x + threadIdx.x;
    if (i >= ET) return;
    const int d = dst[i];
    const float w = expf(ebuf[i] - f32_unorder(mord[d]));
    wbuf[i] = w;
    atomicAdd(&z[d], w);                      // GLOBAL_ATOMIC_ADD_F32
}

__global__ void k_edge_scatter(const float* __restrict__ H, const int* __restrict__ src,
                               const int* __restrict__ dst, const float* __restrict__ wbuf,
                               const float* __restrict__ z, float* __restrict__ out,
                               int ET, int C)
{
    const long long idx = (long long)blockIdx.x * blockDim.x + threadIdx.x;
    if (idx >= (long long)ET * C) return;
    const int e = (int)(idx / C);
    const int c = (int)(idx % C);
    const int s = src[e], d = dst[e];
    const float alpha = wbuf[e] / z[d];
    atomicAdd(&out[(size_t)d * C + c], H[(size_t)s * C + c] * alpha);
}

// ========================= aisle scatter-mean ==============================
__global__ void k_aisle_zero(float* __restrict__ ssum, float* __restrict__ cnt, int S, int O)
{
    const int i = blockIdx.x * blockDim.x + threadIdx.x;
    if (i < S * O) ssum[i] = 0.f;
    if (i < S)     cnt[i]  = 0.f;
}

__global__ void k_aisle_accum(const float* __restrict__ X, const int* __restrict__ aisle,
                              float* __restrict__ ssum, float* __restrict__ cnt,
                              int NT, int N, int NA, int O)
{
    const long long idx = (long long)blockIdx.x * blockDim.x + threadIdx.x;
    if (idx >= (long long)NT * O) return;
    const int n = (int)(idx / O);
    const int c = (int)(idx % O);
    const int aid = aisle[n] + (n / N) * NA;
    atomicAdd(&ssum[(size_t)aid * O + c], X[idx]);
    if (c == 0) atomicAdd(&cnt[aid], 1.0f);
}

__global__ void k_concat(const float* __restrict__ X, const int* __restrict__ aisle,
                         const float* __restrict__ ssum, const float* __restrict__ cnt,
                         float* __restrict__ out, int NT, int N, int NA, int O)
{
    const long long idx = (long long)blockIdx.x * blockDim.x + threadIdx.x;
    const int C2 = 2 * O;
    if (idx >= (long long)NT * C2) return;
    const int n = (int)(idx / C2);
    const int c = (int)(idx % C2);
    float v;
    if (c < O) {
        v = X[(size_t)n * O + c];
    } else {
        const int aid = aisle[n] + (n / N) * NA;
        v = ssum[(size_t)aid * O + (c - O)] / fmaxf(cnt[aid], 1.0f);
    }
    out[idx] = v;
}

// ============================= head (Nc==1) ================================
__global__ void k_head(const float* __restrict__ X, const float* __restrict__ w,
                       const float* __restrict__ b, float* __restrict__ out, int NT, int C)
{
    const int node = blockIdx.x * (blockDim.x >> 5) + ((int)threadIdx.x >> 5);
    const int lane = threadIdx.x & 31;
    if (node >= NT) return;
    float s = 0.f;
    for (int c = lane; c < C; c += 32) s += X[(size_t)node * C + c] * w[c];
    #pragma unroll
    for (int off = 16; off; off >>= 1) s += __shfl_down(s, off, 32);
    if (lane == 0) out[node] = s + b[0];
}

// ========================== masked softmax =================================
__global__ void k_softmax(const float* __restrict__ logits, const int* __restrict__ mask,
                          float* __restrict__ out, int Ncols)
{
    const int b   = blockIdx.x;
    const int tid = threadIdx.x;
    __shared__ float sred[256];
    const float* lg = logits + (size_t)b * Ncols;
    const int*   mk = mask   + (size_t)b * Ncols;

    float m = -3.4e38f;
    for (int i = tid; i < Ncols; i += 256)
        if (mk[i] != 0) m = fmaxf(m, lg[i]);
    sred[tid] = m; __syncthreads();
    for (int s = 128; s; s >>= 1) { if (tid < s) sred[tid] = fmaxf(sred[tid], sred[tid + s]); __syncthreads(); }
    m = sred[0]; __syncthreads();

    float sum = 0.f;
    for (int i = tid; i < Ncols; i += 256)
        if (mk[i] != 0) sum += expf(lg[i] - m);
    sred[tid] = sum; __syncthreads();
    for (int s = 128; s; s >>= 1) { if (tid < s) sred[tid] += sred[tid + s]; __syncthreads(); }
    sum = sred[0];

    const float inv = 1.0f / sum;
    for (int i = tid; i < Ncols; i += 256)
        out[(size_t)b * Ncols + i] = (mk[i] != 0) ? expf(lg[i] - m) * inv : 0.0f;
}

// ============================== host side ==================================
extern "C" void kernel_launch(void* const* d_in, const int* in_sizes, int n_in,
                              void* d_out, int out_size, void* d_ws, size_t ws_size,
                              hipStream_t stream)
{
    constexpr int B = 16, N = 2048, E = 32768, Fin = 32, Hc = 128, Oc = 64, HA = 256, NA = 20;
    const int NT = B * N;           // 32768 nodes
    const int ET = B * E + NT;      // 557056 edges incl. self loops
    const int S  = B * NA;          // 320 aisle segments

    // ---- inputs (setup_inputs order) ----
    const float* graph_nodes = (const float*)d_in[0];
    const int*   aisle_nrs   = (const int*)  d_in[1];
    const int*   edge_links  = (const int*)  d_in[2];
    const int*   mask        = (const int*)  d_in[3];
    // d_in[4] picks_left: unused by reference
    const float* W1 = (const float*)d_in[5];
    const float* as1 = (const float*)d_in[6];
    const float* ad1 = (const float*)d_in[7];
    const float* b1  = (const float*)d_in[8];
    const float* W2 = (const float*)d_in[9];
    const float* as2 = (const float*)d_in[10];
    const float* ad2 = (const float*)d_in[11];
    const float* b2  = (const float*)d_in[12];
    const float* W3 = (const float*)d_in[13];
    const float* as3 = (const float*)d_in[14];
    const float* ad3 = (const float*)d_in[15];
    const float* b3  = (const float*)d_in[16];
    const float* lw1 = (const float*)d_in[17];
    const float* lb1 = (const float*)d_in[18];
    const float* lw2 = (const float*)d_in[19];
    const float* lb2 = (const float*)d_in[20];
    const float* lw3 = (const float*)d_in[21];
    const float* lb3 = (const float*)d_in[22];
    float* outp = (float*)d_out;

    // ---- workspace carve-up (~75 MB) ----
    char*  wsp = (char*)d_ws;
    size_t cur = 0;
    auto alloc = [&](size_t bytes) -> void* {
        cur = (cur + 255) & ~(size_t)255;
        void* r = wsp + cur;
        cur += bytes;
        return r;
    };
    int*      srcA   = (int*)     alloc((size_t)ET * 4);
    int*      dstA   = (int*)     alloc((size_t)ET * 4);
    float*    hA     = (float*)   alloc((size_t)NT * Hc * 4);   // 16 MB ping
    float*    hB     = (float*)   alloc((size_t)NT * Hc * 4);   // 16 MB pong
    float*    M1     = (float*)   alloc((size_t)NT * HA * 4);   // 32 MB MLP hidden
    float*    es     = (float*)   alloc((size_t)NT * 4);
    float*    edv    = (float*)   alloc((size_t)NT * 4);
    unsigned* mord   = (unsigned*)alloc((size_t)NT * 4);
    float*    zbuf   = (float*)   alloc((size_t)NT * 4);
    float*    lgts   = (float*)   alloc((size_t)NT * 4);
    float*    ebuf   = (float*)   alloc((size_t)ET * 4);
    float*    wbuf   = (float*)   alloc((size_t)ET * 4);
    float*    ssum   = (float*)   alloc((size_t)S * Oc * 4);
    float*    cntb   = (float*)   alloc((size_t)S * 4);
    (void)ws_size; (void)n_in; (void)in_sizes; (void)out_size;

    const int TPB = 256;
    auto cdiv = [](long long a, long long b) { return (int)((a + b - 1) / b); };

    // ---- build batched edge list (+ self loops) ----
    k_build_edges<<<cdiv(ET, TPB), TPB, 0, stream>>>(edge_links, srcA, dstA, B, E, N, ET);

    // ---- one GAT layer ----
    auto gat_layer = [&](const float* X, int Cin, int Cout, const float* W,
                         const float* avs, const float* avd, const float* bias,
                         float* hbuf, float* outbuf) {
        const int tiles = (NT / 16) * (Cout / 16);
        k_gemm_wmma<<<tiles / 8, TPB, 0, stream>>>(X, W, nullptr, hbuf, NT, Cin, Cout, 0.f, 0);
        k_node_dots<<<NT / 8, TPB, 0, stream>>>(hbuf, avs, avd, es, edv, NT, Cout);
        k_init_mz<<<cdiv(NT, TPB), TPB, 0, stream>>>(mord, zbuf, NT);
        k_init_bias<<<cdiv((long long)NT * Cout, TPB), TPB, 0, stream>>>(outbuf, bias, NT, Cout);
        k_edge_scores<<<cdiv(ET, TPB), TPB, 0, stream>>>(es, edv, srcA, dstA, ebuf, mord, ET);
        k_edge_w<<<cdiv(ET, TPB), TPB, 0, stream>>>(ebuf, mord, dstA, wbuf, zbuf, ET);
        k_edge_scatter<<<cdiv((long long)ET * Cout, TPB), TPB, 0, stream>>>(
            hbuf, srcA, dstA, wbuf, zbuf, outbuf, ET, Cout);
    };

    // Layer 1: Fin->H   (in: graph_nodes, h: hA, out: hB)
    gat_layer(graph_nodes, Fin, Hc, W1, as1, ad1, b1, hA, hB);
    // Layer 2: H->H     (in: hB, h: hA, out: hB  -- stream-serialized, safe)
    gat_layer(hB, Hc, Hc, W2, as2, ad2, b2, hA, hB);
    // Layer 3: H->O     (in: hB, h: hA, out: hB)
    gat_layer(hB, Hc, Oc, W3, as3, ad3, b3, hA, hB);

    // ---- aisle scatter-mean + concat [x, emb[aid]] -> hA (NT x 128) ----
    k_aisle_zero<<<cdiv(S * Oc, TPB), TPB, 0, stream>>>(ssum, cntb, S, Oc);
    k_aisle_accum<<<cdiv((long long)NT * Oc, TPB), TPB, 0, stream>>>(
        hB, aisle_nrs, ssum, cntb, NT, N, NA, Oc);
    k_concat<<<cdiv((long long)NT * 2 * Oc, TPB), TPB, 0, stream>>>(
        hB, aisle_nrs, ssum, cntb, hA, NT, N, NA, Oc);

    // ---- MLP head ----
    k_gemm_wmma<<<((NT / 16) * (HA / 16)) / 8, TPB, 0, stream>>>(
        hA, lw1, lb1, M1, NT, 2 * Oc, HA, NEG_SLOPE_MLP, 1);
    k_gemm_wmma<<<((NT / 16) * (Oc / 16)) / 8, TPB, 0, stream>>>(
        M1, lw2, lb2, hB, NT, HA, Oc, NEG_SLOPE_MLP, 1);
    k_head<<<NT / 8, TPB, 0, stream>>>(hB, lw3, lb3, lgts, NT, Oc);

    // ---- masked softmax per batch row ----
    k_softmax<<<B, TPB, 0, stream>>>(lgts, mask, outp, N);
}